// HybridSamplerEstimatorQLSTMGraphNet_65481071409456
// MI455X (gfx1250) — compile-verified
//
#include <hip/hip_runtime.h>

// ---------------------------------------------------------------------------
// HybridSamplerEstimatorQLSTM fused kernel for gfx1250 (MI455X, wave32).
//
// One wave32 handles 16 batch elements. Per time step the four gate
// pre-activations (4 gates x H=4 = 16 outputs) for all 16 batch elements are
// produced by TWO chained V_WMMA_F32_16X16X4_F32 ops:
//   D = A_h(16x4) x hx^T(4x16)  +  A_x(16x4) x [x0,x1,1,0]^T(4x16)
// with theta folded into the bias row of A_x (qlayer computes cos(z+theta)).
// D layout puts gates {f,i} of batch b in lane b and gates {u,o} in lane
// b+16; after the per-lane qlayer + branchless activation, one shfl_xor(16)
// pair gives every lane the full (f,i,g,o) and the cell update runs with
// hx/cx replicated across the lane pair. Four of the five outputs come from this one
// fused pass. Activation selection (sigmoid vs tanh) is done via per-lane
// constants  a = mA*rcp(1+exp(sA*q)) + cA  so there is NO EXEC divergence in
// the serial dependency chain; the next step's x is prefetched before the
// transcendental chain to hide load latency.
// ---------------------------------------------------------------------------

typedef __attribute__((ext_vector_type(2))) float v2f;
typedef __attribute__((ext_vector_type(8))) float v8f;

#define B_SZ  4096
#define S_LEN 512

__device__ __forceinline__ float fast_sigmoid(float x) {
    return __builtin_amdgcn_rcpf(1.0f + __expf(-x));
}
__device__ __forceinline__ float fast_tanh(float x) {
    // tanh(x) = 1 - 2/(exp(2x)+1)
    return 1.0f - 2.0f * __builtin_amdgcn_rcpf(1.0f + __expf(2.0f * x));
}

__global__ void qlstm_fused_kernel(
    const float* __restrict__ inputs,
    const float* __restrict__ W_f, const float* __restrict__ b_f,
    const float* __restrict__ W_i, const float* __restrict__ b_i,
    const float* __restrict__ W_u, const float* __restrict__ b_u,
    const float* __restrict__ W_o, const float* __restrict__ b_o,
    const float* __restrict__ th_f, const float* __restrict__ th_i,
    const float* __restrict__ th_u, const float* __restrict__ th_o,
    const float* __restrict__ samp_w, const float* __restrict__ est_w,
    float* __restrict__ out_sl, float* __restrict__ out_sp,
    float* __restrict__ out_eo, float* __restrict__ out_lo)
{
    const int lane  = threadIdx.x & 31;
    const int wave  = (blockIdx.x * blockDim.x + threadIdx.x) >> 5;
    const int m     = lane & 15;         // A row (gate-output idx) / D column (batch idx)
    const bool hi   = lane >= 16;
    const int batch = wave * 16 + m;

    // ---- loop-invariant A matrices (16x16x4 f32 WMMA layout):
    // A row m = gate*4 + j, gates ordered f,i,u,o.
    // lanes 0-15 hold K=0,1 in {v0,v1}; lanes 16-31 hold K=2,3.
    const int gm = m >> 2;               // gate 0..3
    const int j  = m & 3;                // hidden idx within gate
    const float* Wg = (gm == 0) ? W_f : (gm == 1) ? W_i : (gm == 2) ? W_u : W_o;
    const float* bg = (gm == 0) ? b_f : (gm == 1) ? b_i : (gm == 2) ? b_u : b_o;
    const float* tg = (gm == 0) ? th_f : (gm == 1) ? th_i : (gm == 2) ? th_u : th_o;

    v2f Ah, Ax;
    Ah[0] = hi ? Wg[j * 6 + 4] : Wg[j * 6 + 2];   // hidden-part weights
    Ah[1] = hi ? Wg[j * 6 + 5] : Wg[j * 6 + 3];
    Ax[0] = hi ? (bg[j] + tg[j]) : Wg[j * 6 + 0]; // input weights / folded bias+theta
    Ax[1] = hi ? 0.0f            : Wg[j * 6 + 1];

    // ---- branchless activation constants:
    // lo lanes (gates f,i):  sigmoid(q) =  1*rcp(1+exp(-1*q)) + 0
    // hi lanes (gate  u  ):  tanh(q)    = -2*rcp(1+exp( 2*q)) + 1
    const float sA = hi ?  2.0f : -1.0f;
    const float mA = hi ? -2.0f :  1.0f;
    const float cA = hi ?  1.0f :  0.0f;

    // ---- sampler / estimator loop-invariant scalars
    const float thsum = samp_w[0] + samp_w[1] + samp_w[2] + samp_w[3];
    const float ct = __cosf(0.5f * thsum), st = __sinf(0.5f * thsum);
    const float sinw = __sinf(est_w[0]);

    // ---- LSTM state (replicated across the (b, b+16) lane pair)
    float hx0 = 0.f, hx1 = 0.f, hx2 = 0.f, hx3 = 0.f;
    float cx0 = 0.f, cx1 = 0.f, cx2 = 0.f, cx3 = 0.f;

    const float2* xin = (const float2*)inputs + (size_t)batch * S_LEN; // (S,2) row
    float2 xv = xin[0];                                                // prefetched x[s]

    for (int s = 0; s < S_LEN; ++s) {
        // ---- B matrices (4x16): K striped across lane halves, N = lane&15
        v2f Bh, Bx;
        Bh[0] = hi ? hx2  : hx0;
        Bh[1] = hi ? hx3  : hx1;
        Bx[0] = hi ? 1.0f : xv.x;
        Bx[1] = hi ? 0.0f : xv.y;

        // ---- z = W_h @ hx + (W_x @ x + b + theta), full f32 via WMMA
        v8f acc = {};
        acc = __builtin_amdgcn_wmma_f32_16x16x4_f32(
                  false, Ah, false, Bh, (short)0, acc, false, false);
        acc = __builtin_amdgcn_wmma_f32_16x16x4_f32(
                  false, Ax, false, Bx, (short)0, acc, false, false);
        // lane b   (lo): acc[0..3]=z_f, acc[4..7]=z_i
        // lane b+16(hi): acc[0..3]=z_u, acc[4..7]=z_o

        // ---- prefetch next step's x while the transcendental chain runs
        const int sn = (s + 1 < S_LEN) ? (s + 1) : s;
        xv = xin[sn];

        // ---- qlayer: c=cos(z); out=[c1c2c3, c0c1, c0c1c2, c0c1c2c3]
        float c0 = __cosf(acc[0]), c1 = __cosf(acc[1]);
        float c2 = __cosf(acc[2]), c3 = __cosf(acc[3]);
        float qa1 = c0 * c1, qa2 = qa1 * c2, qa3 = qa2 * c3, qa0 = c1 * c2 * c3;
        float d0 = __cosf(acc[4]), d1 = __cosf(acc[5]);
        float d2 = __cosf(acc[6]), d3 = __cosf(acc[7]);
        float qb1 = d0 * d1, qb2 = qb1 * d2, qb3 = qb2 * d3, qb0 = d1 * d2 * d3;

        // ---- branchless activations: lo->sigmoid (f), hi->tanh (u=g)
        float a0 = fmaf(mA, __builtin_amdgcn_rcpf(1.0f + __expf(sA * qa0)), cA);
        float a1 = fmaf(mA, __builtin_amdgcn_rcpf(1.0f + __expf(sA * qa1)), cA);
        float a2 = fmaf(mA, __builtin_amdgcn_rcpf(1.0f + __expf(sA * qa2)), cA);
        float a3 = fmaf(mA, __builtin_amdgcn_rcpf(1.0f + __expf(sA * qa3)), cA);
        // second gate in each half (i / o) is always sigmoid
        float e0 = fast_sigmoid(qb0), e1 = fast_sigmoid(qb1);
        float e2 = fast_sigmoid(qb2), e3 = fast_sigmoid(qb3);

        // ---- exchange halves (uniform flow; wave32 xor-16 swap)
        float oa0 = __shfl_xor(a0, 16, 32), oa1 = __shfl_xor(a1, 16, 32);
        float oa2 = __shfl_xor(a2, 16, 32), oa3 = __shfl_xor(a3, 16, 32);
        float ob0 = __shfl_xor(e0, 16, 32), ob1 = __shfl_xor(e1, 16, 32);
        float ob2 = __shfl_xor(e2, 16, 32), ob3 = __shfl_xor(e3, 16, 32);

        float f0 = hi ? oa0 : a0, f1 = hi ? oa1 : a1, f2 = hi ? oa2 : a2, f3 = hi ? oa3 : a3;
        float g0 = hi ? a0 : oa0, g1 = hi ? a1 : oa1, g2 = hi ? a2 : oa2, g3 = hi ? a3 : oa3;
        float i0 = hi ? ob0 : e0, i1 = hi ? ob1 : e1, i2 = hi ? ob2 : e2, i3 = hi ? ob3 : e3;
        float o0 = hi ? e0 : ob0, o1 = hi ? e1 : ob1, o2 = hi ? e2 : ob2, o3 = hi ? e3 : ob3;

        // ---- cell update (replicated across the lane pair; tanh is uniform)
        cx0 = f0 * cx0 + i0 * g0;  cx1 = f1 * cx1 + i1 * g1;
        cx2 = f2 * cx2 + i2 * g2;  cx3 = f3 * cx3 + i3 * g3;
        hx0 = o0 * fast_tanh(cx0); hx1 = o1 * fast_tanh(cx1);
        hx2 = o2 * fast_tanh(cx2); hx3 = o3 * fast_tanh(cx3);

        // ---- outputs for this (batch, s); lo/hi lanes write disjoint tensors
        const size_t bs = (size_t)batch * S_LEN + s;
        if (!hi) {
            float4 h4 = make_float4(hx0, hx1, hx2, hx3);
            *(float4*)(out_lo + bs * 4) = h4;                 // lstm_out
            out_eo[bs] = sinw * __sinf(hx0);                  // estimator <Y>
        } else {
            const float p0 = hx0, p1 = hx1;
            const float ca = __cosf(0.5f * p0), sa = __sinf(0.5f * p0);
            const float cb = __cosf(0.5f * p1), sb = __sinf(0.5f * p1);
            const float C00 = ct * cb * ca - st * sb * sa;
            const float C10 = ct * sb * ca - st * cb * sa;
            const float P0  = C00 * C00 + C10 * C10;
            float2 sl = make_float2(P0, 1.0f - P0);
            *(float2*)(out_sl + bs * 2) = sl;                 // sampler logits
            const float sp0 = fast_sigmoid(2.0f * P0 - 1.0f); // softmax of 2
            float2 sp = make_float2(sp0, 1.0f - sp0);
            *(float2*)(out_sp + bs * 2) = sp;                 // sampler probs
        }
    }
}

__global__ void fid_adj_kernel(float* __restrict__ out) {
    const int idx = blockIdx.x * blockDim.x + threadIdx.x;
    if (idx < S_LEN * S_LEN) {
        const int r = idx / S_LEN, c = idx - r * S_LEN;
        out[idx] = (r == c) ? 0.0f : 1.0f;
    }
}

extern "C" void kernel_launch(void* const* d_in, const int* in_sizes, int n_in,
                              void* d_out, int out_size, void* d_ws, size_t ws_size,
                              hipStream_t stream) {
    (void)in_sizes; (void)n_in; (void)out_size; (void)d_ws; (void)ws_size;
    const float* inputs = (const float*)d_in[0];
    const float* W_f = (const float*)d_in[1];
    const float* b_f = (const float*)d_in[2];
    const float* W_i = (const float*)d_in[3];
    const float* b_i = (const float*)d_in[4];
    const float* W_u = (const float*)d_in[5];
    const float* b_u = (const float*)d_in[6];
    const float* W_o = (const float*)d_in[7];
    const float* b_o = (const float*)d_in[8];
    const float* th_f = (const float*)d_in[9];
    const float* th_i = (const float*)d_in[10];
    const float* th_u = (const float*)d_in[11];
    const float* th_o = (const float*)d_in[12];
    const float* samp = (const float*)d_in[13];
    const float* estw = (const float*)d_in[14];

    float* out    = (float*)d_out;
    float* out_sl = out;                                      // (B,S,2)
    float* out_sp = out_sl + (size_t)B_SZ * S_LEN * 2;        // (B,S,2)
    float* out_eo = out_sp + (size_t)B_SZ * S_LEN * 2;        // (B,S,1)
    float* out_lo = out_eo + (size_t)B_SZ * S_LEN;            // (B,S,4)
    float* out_fd = out_lo + (size_t)B_SZ * S_LEN * 4;        // (S,S)

    // 16 batch elems per wave32 -> 256 waves -> 8192 threads
    const int threads = (B_SZ / 16) * 32;
    const int block = 128;                                    // 4 waves/block
    qlstm_fused_kernel<<<threads / block, block, 0, stream>>>(
        inputs, W_f, b_f, W_i, b_i, W_u, b_u, W_o, b_o,
        th_f, th_i, th_u, th_o, samp, estw,
        out_sl, out_sp, out_eo, out_lo);

    fid_adj_kernel<<<(S_LEN * S_LEN + 255) / 256, 256, 0, stream>>>(out_fd);
}